// Model_1735166788403
// MI455X (gfx1250) — compile-verified
//
#include <hip/hip_runtime.h>

typedef __attribute__((ext_vector_type(2))) float v2f;
typedef __attribute__((ext_vector_type(8))) float v8f;

#define OC 16          // output channels
#define KTOT 27        // 3*3*3 reduction
#define PIX 900        // 30*30 output pixels per image
#define TILES_PER_IMG 57   // ceil(900/16)
#define TILES_PER_WAVE 4
#define WAVES_PER_BLOCK 8  // 256 threads

__global__ __launch_bounds__(256)
void conv3x3_hswish_wmma(const float* __restrict__ x,
                         const float* __restrict__ w,
                         const float* __restrict__ bias,
                         float* __restrict__ out,
                         int total_tiles)
{
    const int lane = threadIdx.x & 31;
    const int wave = blockIdx.x * WAVES_PER_BLOCK + (threadIdx.x >> 5);
    const int half = lane >> 4;    // 0: lanes 0-15, 1: lanes 16-31
    const int l16  = lane & 15;

    // ---- A fragments (weights, OIHW flat = co*27 + k), loaded once per wave.
    // A is 16(co) x 4(k) per WMMA: lane l16 = co, VGPR i holds k = 4t + i + 2*half.
    v2f afrag[7];
    #pragma unroll
    for (int t = 0; t < 7; ++t) {
        #pragma unroll
        for (int i = 0; i < 2; ++i) {
            int k  = 4 * t + i + 2 * half;
            int kk = (k < KTOT) ? k : 0;
            float v = w[l16 * KTOT + kk];
            afrag[t][i] = (k < KTOT) ? v : 0.0f;
        }
    }
    // Bias for this lane's 8 output rows: co = r + 8*half.
    float bvals[8];
    #pragma unroll
    for (int r = 0; r < 8; ++r) bvals[r] = bias[r + 8 * half];

    #pragma unroll 1
    for (int j = 0; j < TILES_PER_WAVE; ++j) {
        int tile = wave * TILES_PER_WAVE + j;
        if (tile >= total_tiles) break;          // wave-uniform: EXEC stays full

        int n     = tile / TILES_PER_IMG;
        int pt    = tile - n * TILES_PER_IMG;
        int pbase = pt * 16;

        int  p  = pbase + l16;
        int  pc = (p < PIX) ? p : (PIX - 1);     // clamp so loads stay in-bounds
        int  oh = pc / 30;
        int  ow = pc - oh * 30;
        const float* xb = x + (size_t)n * 3072 + oh * 32 + ow;

        // Prefetch next tile's input rows (speculative, gfx1250 global_prefetch_b8).
        if (j + 1 < TILES_PER_WAVE) {
            __builtin_prefetch((const void*)(xb + 16), 0, 0);
        }

        // ---- B fragments + 7 chained f32 WMMAs (K = 28, last slot zero).
        v8f acc = {};
        #pragma unroll
        for (int t = 0; t < 7; ++t) {
            v2f bfrag;
            #pragma unroll
            for (int i = 0; i < 2; ++i) {
                int  k  = 4 * t + i + 2 * half;
                bool kv = (k < KTOT);
                int  kk = kv ? k : 0;
                int  ci = kk / 9;
                int  r9 = kk - ci * 9;
                int  kh = r9 / 3;
                int  kw = r9 - kh * 3;
                float v = xb[ci * 1024 + kh * 32 + kw];
                bfrag[i] = kv ? v : 0.0f;
            }
            // 8 args: (neg_a, A, neg_b, B, c_mod, C, reuse_a, reuse_b)
            acc = __builtin_amdgcn_wmma_f32_16x16x4_f32(
                false, afrag[t], false, bfrag, (short)0, acc, false, false);
        }

        // ---- Epilogue: bias + relu(v)*min(v+3,6)/6, coalesced predicated stores.
        bool ok = (p < PIX);
        float* ob = out + ((size_t)n * OC + (size_t)(8 * half)) * PIX + p;
        #pragma unroll
        for (int r = 0; r < 8; ++r) {
            float v   = acc[r] + bvals[r];
            float res = fmaxf(v, 0.0f) * fminf(v + 3.0f, 6.0f) * (1.0f / 6.0f);
            if (ok) ob[(size_t)r * PIX] = res;
        }
    }
}

extern "C" void kernel_launch(void* const* d_in, const int* in_sizes, int n_in,
                              void* d_out, int out_size, void* d_ws, size_t ws_size,
                              hipStream_t stream) {
    const float* x    = (const float*)d_in[0];   // (N,3,32,32) f32
    const float* wgt  = (const float*)d_in[1];   // (16,3,3,3)  f32
    const float* bias = (const float*)d_in[2];   // (16,)       f32
    float* out = (float*)d_out;                  // (N,16,30,30) f32

    int N = in_sizes[0] / 3072;                  // 3*32*32
    int total_tiles = N * TILES_PER_IMG;
    int waves  = (total_tiles + TILES_PER_WAVE - 1) / TILES_PER_WAVE;
    int blocks = (waves + WAVES_PER_BLOCK - 1) / WAVES_PER_BLOCK;

    conv3x3_hswish_wmma<<<blocks, 256, 0, stream>>>(x, wgt, bias, out, total_tiles);
}